// CoordinateDescent_90443421319620
// MI455X (gfx1250) — compile-verified
//
#include <hip/hip_runtime.h>
#include <hip/hip_bf16.h>

// ---------------------------------------------------------------------------
// Coordinate-descent NMF step for MI455X (gfx1250, wave32).
// B=128, M=N=512, R=16, l1=l2=0, eps=1e-16.
//
// HBM-bound: x (128MB) is streamed exactly twice (x@v, xT@u) -> ~11us floor at
// 23.3 TB/s. Matmuls use V_WMMA_F32_16X16X4_F32 (fp32, no conversion cost).
// The x@v pass stages x through LDS with GLOBAL_LOAD_ASYNC_TO_LDS_B128
// (double-buffered, ASYNCcnt-tracked) so every HBM transaction is a coalesced
// 128B burst. w-statistics are factorized so z = u v^T is never materialized.
// ---------------------------------------------------------------------------

#define B_   128
#define M_   512
#define N_   512
#define R_   16
#define EPSF 1e-16f

typedef __attribute__((ext_vector_type(2))) float v2f;
typedef __attribute__((ext_vector_type(8))) float v8f;
typedef int v4i_ __attribute__((vector_size(16)));   // matches builtin's V4i param

// V_WMMA_F32_16X16X4_F32 (codegen-confirmed in round 1).
#if defined(__gfx1250__) && __has_builtin(__builtin_amdgcn_wmma_f32_16x16x4_f32)
#define WMMA4(c, a, bb) \
  (c) = __builtin_amdgcn_wmma_f32_16x16x4_f32(false, (a), false, (bb), (short)0, (c), false, false)
#else
// host-pass / fallback so the file always parses; device path is the WMMA above
#define WMMA4(c, a, bb) do { (c)[0] += (a).x * (bb).x; (c)[1] += (a).y * (bb).y; } while (0)
#endif

// CDNA5 async global->LDS engine (ASYNCcnt). Signature discovered from round-2
// diagnostic: (v4i addrspace(1)*, v4i addrspace(3)*, imm offset, imm cpol).
#if defined(__gfx1250__) && \
    __has_builtin(__builtin_amdgcn_global_load_async_to_lds_b128) && \
    __has_builtin(__builtin_amdgcn_s_wait_asynccnt)
#define HAVE_ASYNC_LDS 1
#endif

// Per-batch scalars: s0 = 1/(w1+eps), w0.
__global__ void k_init(const float* __restrict__ w, float* __restrict__ s0w0) {
  int b = threadIdx.x;
  if (b < B_) {
    float w0 = w[b * 2 + 0];
    float w1 = w[b * 2 + 1];
    s0w0[b]      = 1.0f / (w1 + EPSF);
    s0w0[B_ + b] = w0;
  }
}

// Gram: G = P^T P (16x16) and colsum(P) per batch; P is (512 x 16) row-major.
// One wave per batch; A-tile and B-tile of a Gram product load identical
// values under the ISA 16x4-f32 layout -> one load pair feeds both operands.
__global__ void k_gram(const float* __restrict__ P,
                       float* __restrict__ G, float* __restrict__ cs) {
  int b    = blockIdx.x;
  int lane = threadIdx.x;          // 0..31, EXEC all ones (WMMA requirement)
  int r    = lane & 15;
  int kh   = (lane >> 4) << 1;     // 0 or 2
  const float* Pb = P + (size_t)b * (512 * 16);
  v8f c = {0.f, 0.f, 0.f, 0.f, 0.f, 0.f, 0.f, 0.f};
  float csum = 0.f;
  for (int k0 = 0; k0 < 512; k0 += 4) {
    float p0 = Pb[(k0 + kh)     * 16 + r];
    float p1 = Pb[(k0 + kh + 1) * 16 + r];
    csum += p0 + p1;                       // each element visited exactly once
    v2f ab = {p0, p1};
    WMMA4(c, ab, ab);
  }
  csum += __shfl_xor(csum, 16, 32);        // fold lanes 16..31 onto 0..15
  if (lane < 16) cs[b * 16 + lane] = csum;
  float* Gb = G + (size_t)b * 256;
  int rowbase = (lane >> 4) << 3;          // C layout: VGPR i -> M=i / M=i+8
  #pragma unroll
  for (int i = 0; i < 8; ++i)
    Gb[(rowbase + i) * 16 + r] = c[i];
}

// GEMM: out[b] = x[b] @ P[b]    (transposeX == 0)  [LDS-staged, async]
//       out[b] = x[b]^T @ P[b]  (transposeX == 1)  [direct, already coalesced]
// One wave per 16x16 output tile (grid = B * 32), K = 512.
#define CHUNK 32          // K-chunk staged per buffer
#define LROW  33          // padded LDS row stride (floats) -> no bank conflicts
__global__ void k_xgemm(const float* __restrict__ x, const float* __restrict__ P,
                        float* __restrict__ out, float* __restrict__ stripsum,
                        int transposeX) {
  __shared__ float xs[2][16 * LROW];
  int b     = blockIdx.x >> 5;
  int strip = blockIdx.x & 31;
  int m0    = strip << 4;
  int lane  = threadIdx.x;
  int r     = lane & 15;
  int kh    = (lane >> 4) << 1;
  const float* xb = x + (size_t)b * (512 * 512);
  const float* Pb = P + (size_t)b * (512 * 16);
  v8f c = {0.f, 0.f, 0.f, 0.f, 0.f, 0.f, 0.f, 0.f};
  float sa = 0.f;

  if (!transposeX) {
    // Coalesced staging: 16 rows x 32 cols of x per chunk; each lane moves
    // 4x float4; each 8-lane group covers one contiguous 128B burst.
    int srow  = lane >> 3;          // 0..3
    int scol4 = (lane & 7) << 2;    // 0,4,...,28
    auto stage = [&](int buf, int kb) {
      #pragma unroll
      for (int j = 0; j < 4; ++j) {
        int row = srow + (j << 2);
        const float* gsrc = xb + (size_t)(m0 + row) * 512 + kb + scol4;
        float* lds = &xs[buf][row * LROW + scol4];
#if defined(HAVE_ASYNC_LDS)
        __builtin_amdgcn_global_load_async_to_lds_b128(
            (__attribute__((address_space(1))) v4i_*)(void*)gsrc,
            (__attribute__((address_space(3))) v4i_*)(void*)lds, 0, 0);
#else
        float4 t = *(const float4*)gsrc;
        *(float4*)lds = t;
#endif
      }
    };

    stage(0, 0);
    for (int chunk = 0; chunk < 512 / CHUNK; ++chunk) {
      int kb  = chunk * CHUNK;
      int buf = chunk & 1;
      if (chunk + 1 < 512 / CHUNK) {
        stage(buf ^ 1, kb + CHUNK);              // prefetch next chunk
#if defined(HAVE_ASYNC_LDS)
        __builtin_amdgcn_s_wait_asynccnt(4);     // current chunk landed
#endif
      } else {
#if defined(HAVE_ASYNC_LDS)
        __builtin_amdgcn_s_wait_asynccnt(0);
#endif
      }
      __syncthreads();
      #pragma unroll
      for (int kk = 0; kk < CHUNK; kk += 4) {
        float a0 = xs[buf][r * LROW + kk + kh];
        float a1 = xs[buf][r * LROW + kk + kh + 1];
        sa += a0 + a1;                           // each x element hit once
        float b0 = Pb[(kb + kk + kh)     * 16 + r];
        float b1 = Pb[(kb + kk + kh + 1) * 16 + r];
        v2f av = {a0, a1};
        v2f bv = {b0, b1};
        WMMA4(c, av, bv);
      }
      __syncthreads();
    }
    if (stripsum) {                              // deterministic sum(x) partials
      #pragma unroll
      for (int off = 16; off > 0; off >>= 1) sa += __shfl_xor(sa, off, 32);
      if (lane == 0) stripsum[b * 32 + strip] = sa;
    }
  } else {
    // x^T path: lanes 0..15 read consecutive columns -> coalesced as-is.
    for (int k0 = 0; k0 < 512; k0 += 4) {
      const float* src = xb + (size_t)(k0 + kh) * 512 + (m0 + r);
      float a0 = src[0];
      float a1 = src[512];
      __builtin_prefetch(src + 64 * 512);        // global_prefetch_b8
      float b0 = Pb[(k0 + kh)     * 16 + r];
      float b1 = Pb[(k0 + kh + 1) * 16 + r];
      v2f av = {a0, a1};
      v2f bv = {b0, b1};
      WMMA4(c, av, bv);
    }
  }

  float* ob = out + (size_t)b * (512 * 16) + (size_t)m0 * 16;
  int rowbase = (lane >> 4) << 3;
  #pragma unroll
  for (int i = 0; i < 8; ++i)
    ob[(rowbase + i) * 16 + r] = c[i];
}

// Coordinate descent over R=16 components, one thread per matrix row.
// `io` holds the RAW GEMM result on entry, the updated factor on exit
// (row-private read-modify-write -> race-free).
// a[r] = s0*(raw[r] - w0*colsum[r]). If szb_part != nullptr, also emits
// per-block partials of <p_new, raw> (= szb, since raw = xT @ u_new).
__global__ void k_cd(const float* __restrict__ G, const float* __restrict__ cs,
                     const float* __restrict__ s0w0, const float* __restrict__ pin,
                     float* __restrict__ io, float* __restrict__ szb_part) {
  __shared__ float Gs[256];
  __shared__ float cs_s[16];
  __shared__ float red[256];
  int b    = blockIdx.x >> 1;
  int half = blockIdx.x & 1;
  int tid  = threadIdx.x;          // 0..255
  int m    = (half << 8) + tid;
  Gs[tid] = G[(size_t)b * 256 + tid];
  if (tid < 16) cs_s[tid] = cs[b * 16 + tid];
  __syncthreads();
  float s0 = s0w0[b], w0 = s0w0[B_ + b];
  float* base     = io  + ((size_t)b * 512 + m) * 16;
  const float* pb = pin + ((size_t)b * 512 + m) * 16;
  float raw[16], a[16], uu[16];
  #pragma unroll
  for (int r = 0; r < 16; ++r) {
    raw[r] = base[r];
    a[r]   = s0 * (raw[r] - w0 * cs_s[r]);
    uu[r]  = pb[r];
  }
  #pragma unroll
  for (int r = 0; r < 16; ++r) {
    float brr = Gs[r * 16 + r];
    float dot = 0.f;
    #pragma unroll
    for (int s = 0; s < 16; ++s) dot = fmaf(uu[s], Gs[s * 16 + r], dot);
    float num = a[r] - (dot - uu[r] * brr);   // soft-threshold(t=0) = identity
    uu[r] = (num + EPSF) / (brr + EPSF);      // l2 = 0
  }
  #pragma unroll
  for (int r = 0; r < 16; ++r) base[r] = uu[r];
  if (szb_part) {                             // uniform branch across block
    float p = 0.f;
    #pragma unroll
    for (int r = 0; r < 16; ++r) p = fmaf(uu[r], raw[r], p);
    red[tid] = p;
    __syncthreads();
    for (int off = 128; off > 0; off >>= 1) {
      if (tid < off) red[tid] += red[tid + off];
      __syncthreads();
    }
    if (tid == 0) szb_part[b * 2 + half] = red[0];
  }
}

// Closed-form 2x2 solve per batch:
//   AtA = [[mn, sz],[sz, szz]], Atb = [sb, szb]
//   sz  = <colsum(u), colsum(v)>,  szz = <u^T u, v^T v>_F,
//   szb = <v_new, x^T u_new>,      sb  = sum(x).
__global__ void k_wfin(const float* __restrict__ stripsum,
                       const float* __restrict__ csu, const float* __restrict__ csv,
                       const float* __restrict__ Gu,  const float* __restrict__ Gv,
                       const float* __restrict__ szb_part, float* __restrict__ wout) {
  int b = threadIdx.x;
  if (b >= B_) return;
  float sb = 0.f;
  for (int i = 0; i < 32; ++i) sb += stripsum[b * 32 + i];
  float sz = 0.f;
  #pragma unroll
  for (int r = 0; r < 16; ++r) sz = fmaf(csu[b * 16 + r], csv[b * 16 + r], sz);
  float szz = 0.f;
  for (int i = 0; i < 256; ++i)
    szz = fmaf(Gu[(size_t)b * 256 + i], Gv[(size_t)b * 256 + i], szz);
  float szb = szb_part[b * 2] + szb_part[b * 2 + 1];
  const float mn = (float)(M_ * N_);
  float det = mn * szz - sz * sz;
  float inv = 1.0f / det;
  wout[b * 2 + 0] = (szz * sb - sz * szb) * inv;
  wout[b * 2 + 1] = (mn * szb - sz * sb) * inv;
}

extern "C" void kernel_launch(void* const* d_in, const int* in_sizes, int n_in,
                              void* d_out, int out_size, void* d_ws, size_t ws_size,
                              hipStream_t stream) {
  const float* x = (const float*)d_in[0];   // (B, M, N)
  const float* u = (const float*)d_in[1];   // (B, M, R)
  const float* v = (const float*)d_in[2];   // (B, N, R)
  const float* w = (const float*)d_in[3];   // (B, 2, 1)

  float* out = (float*)d_out;               // [u | v | w] flat
  float* uo  = out;                                  // B*M*R
  float* vo  = out + (size_t)B_ * M_ * R_;           // B*N*R
  float* wo  = out + 2 * (size_t)B_ * M_ * R_;       // B*2

  // Workspace (small arrays only; raw GEMM results live in d_out regions)
  float* ws      = (float*)d_ws;
  float* s0w0    = ws;                  // 2*B
  float* Gv_old  = s0w0   + 2 * B_;     // B*256
  float* csv_old = Gv_old + B_ * 256;   // B*16
  float* Gu_new  = csv_old + B_ * 16;   // B*256
  float* csu_new = Gu_new + B_ * 256;   // B*16
  float* Gv_new  = csu_new + B_ * 16;   // B*256
  float* csv_new = Gv_new + B_ * 256;   // B*16
  float* strips  = csv_new + B_ * 16;   // B*32
  float* szbp    = strips + B_ * 32;    // B*2

  // 1) scalars
  k_init<<<1, 128, 0, stream>>>(w, s0w0);
  // 2) b = v^T v and colsum(v) (old v)
  k_gram<<<B_, 32, 0, stream>>>(v, Gv_old, csv_old);
  // 3) raw x@v -> u region; strip sums of x (for sb)
  k_xgemm<<<B_ * 32, 32, 0, stream>>>(x, v, uo, strips, 0);
  // 4) u coordinate descent (in-place on u region)
  k_cd<<<B_ * 2, 256, 0, stream>>>(Gv_old, csv_old, s0w0, u, uo, nullptr);
  // 5) b = u_new^T u_new and colsum(u_new)
  k_gram<<<B_, 32, 0, stream>>>(uo, Gu_new, csu_new);
  // 6) raw x^T@u_new -> v region
  k_xgemm<<<B_ * 32, 32, 0, stream>>>(x, uo, vo, nullptr, 1);
  // 7) v coordinate descent (in-place on v region) + szb partials
  k_cd<<<B_ * 2, 256, 0, stream>>>(Gu_new, csu_new, s0w0, v, vo, szbp);
  // 8) v_new^T v_new and colsum(v_new)
  k_gram<<<B_, 32, 0, stream>>>(vo, Gv_new, csv_new);
  // 9) w = solve(AtA, Atb)
  k_wfin<<<1, 128, 0, stream>>>(strips, csu_new, csv_new, Gu_new, Gv_new, szbp, wo);
}